// GNNTwoLayer_79422535238248
// MI455X (gfx1250) — compile-verified
//
#include <hip/hip_runtime.h>
#include <hip/hip_bf16.h>
#include <math.h>

typedef __attribute__((ext_vector_type(2))) float v2f;
typedef __attribute__((ext_vector_type(8))) float v8f;
typedef __attribute__((ext_vector_type(4))) unsigned int v4u;
typedef __attribute__((ext_vector_type(8))) int v8i_t;
typedef __attribute__((ext_vector_type(4))) int v4i_t;

#define FEAT 128
#define EPS 1e-5f

#if defined(__has_builtin)
#if __has_builtin(__builtin_amdgcn_tensor_load_to_lds)
#define HAVE_TDM 1
#endif
#endif

// ---------------------------------------------------------------------------
// Degree kernels: deg[i] = 1 (self loop) + sum_e w[e] for dst[e]==i ; then
// dinv[i] = rsqrt(deg[i]) (deg>0 guaranteed by the self loop, matching ref).
// ---------------------------------------------------------------------------
__global__ void __launch_bounds__(256) deg_init_kernel(float* __restrict__ deg, int n) {
    int i = blockIdx.x * 256 + threadIdx.x;
    if (i < n) deg[i] = 1.0f;               // self-loop weight
}

__global__ void __launch_bounds__(256) deg_scatter_kernel(const long long* __restrict__ dst,
                                                          const float* __restrict__ w,
                                                          float* __restrict__ deg, int e) {
    int i = blockIdx.x * 256 + threadIdx.x;
    if (i < e) atomicAdd(&deg[dst[i]], w[i]);
}

__global__ void __launch_bounds__(256) deg_finalize_kernel(float* __restrict__ deg, int n) {
    int i = blockIdx.x * 256 + threadIdx.x;
    if (i < n) {
        float d = deg[i];
        deg[i] = (d > 0.0f) ? rsqrtf(d) : 0.0f;
    }
}

// ---------------------------------------------------------------------------
// GEMM: H[N,128] = X[N,128] @ W[128,128] via V_WMMA_F32_16X16X4_F32.
// Block = 256 threads = 8 waves; block covers 16 rows x 128 cols (one 16x16
// tile per wave). W (64 KB) is staged once per block into LDS — by the
// Tensor Data Mover (TENSOR_LOAD_TO_LDS + s_wait_tensorcnt), issued by
// wave 0 with the D# built per ISA §8 — and reused by all 8 waves across the
// 32-step K loop. N = 100000 = 6250*16, no tail, EXEC all-ones for WMMA.
//
// Fragment layouts (ISA 7.12.2, 32-bit elements, wave32):
//   A 16x4 : lane<16 -> M=lane,    VGPR{0,1}=K{0,1}; lane>=16 -> K{2,3}
//   B 4x16 : VGPR v, lane<16 -> K=v,   N=lane; lane>=16 -> K=v+2, N=lane-16
//   C 16x16: VGPR v, lane<16 -> M=v,   N=lane; lane>=16 -> M=v+8, N=lane-16
// ---------------------------------------------------------------------------
__global__ void __launch_bounds__(256)
gemm128_wmma_kernel(const float* __restrict__ X, const float* __restrict__ W,
                    float* __restrict__ H) {
    __shared__ float Ws[FEAT * FEAT];   // 64 KB of the 320 KB WGP LDS
    const int tid = threadIdx.x;

#ifdef HAVE_TDM
    if (tid < 32) {   // wave 0 issues one TDM descriptor for the whole tile
        unsigned ldsaddr = (unsigned)(size_t)(void*)Ws;              // LDS offset
        unsigned long long ga = (unsigned long long)(size_t)(const void*)W;
        v4u g0;
        g0.x = 1u;                                                   // count=1, user D#
        g0.y = ldsaddr;                                              // lds_addr
        g0.z = (unsigned)(ga & 0xFFFFFFFFu);                         // global_addr[31:0]
        g0.w = (unsigned)((ga >> 32) & 0x1FFFFFFu) | 0x80000000u;    // addr[56:32] | type=2
        // group1: data_size=4B (2<<16); tensor_dim0=128 ([79:48]);
        // tensor_dim1=128 ([111:80]); tile_dim0=128 ([127:112]);
        // tile_dim1=128 ([143:128]); tensor_dim0_stride=128 ([207:160])
        v8i_t g1 = { 0x00020000, 0x00800000, 0x00800000, 0x00800000, 128, 128, 0, 0 };
        v4i_t g2 = { 0, 0, 0, 0 };
        v4i_t g3 = { 0, 0, 0, 0 };
        v8i_t g4 = { 0, 0, 0, 0, 0, 0, 0, 0 };
        // 6-arg form on this toolchain: (g0, g1, g2, g3, g4, cpol)
        __builtin_amdgcn_tensor_load_to_lds(g0, g1, g2, g3, g4, 0);
#if __has_builtin(__builtin_amdgcn_s_wait_tensorcnt)
        __builtin_amdgcn_s_wait_tensorcnt(0);
#else
        asm volatile("s_wait_tensorcnt 0" ::: "memory");
#endif
    }
    __syncthreads();
#else
    for (int i = tid; i < FEAT * FEAT; i += 256) Ws[i] = W[i];
    __syncthreads();
#endif

    const int lane  = tid & 31;
    const int wave  = tid >> 5;
    const int row0  = blockIdx.x * 16;
    const int col0  = wave * 16;
    const int m     = lane & 15;          // row within tile (A) / col (B,C)
    const int khalf = (lane >> 4) * 2;    // 0 for lanes 0-15, 2 for 16-31

    const float* xrow = X + (size_t)(row0 + m) * FEAT;

    v8f acc = {};
    for (int k = 0; k < FEAT; k += 4) {
        // A fragment: two consecutive K values for this lane's row
        v2f a = *(const v2f*)(xrow + k + khalf);
        // B fragment: rows k+khalf, k+khalf+1 of W at this lane's column
        v2f b;
        b.x = Ws[(k + khalf + 0) * FEAT + col0 + m];
        b.y = Ws[(k + khalf + 1) * FEAT + col0 + m];
        acc = __builtin_amdgcn_wmma_f32_16x16x4_f32(
            /*neg_a=*/false, a, /*neg_b=*/false, b,
            /*c_mod=*/(short)0, acc, /*reuse_a=*/false, /*reuse_b=*/false);
    }

    const int col   = col0 + m;
    const int rbase = row0 + (lane >> 4) * 8;
#pragma unroll
    for (int v = 0; v < 8; ++v)
        H[(size_t)(rbase + v) * FEAT + col] = acc[v];
}

// ---------------------------------------------------------------------------
// AGG[i][f] = b[f] + H[i][f] * dinv[i]^2   (bias + self-loop contribution)
// i is wave-uniform (128 feats = 4 waves/node): readfirstlane -> scalar load.
// ---------------------------------------------------------------------------
__global__ void __launch_bounds__(256)
agg_init_kernel(const float* __restrict__ H, const float* __restrict__ dinv,
                const float* __restrict__ b, float* __restrict__ AGG, int n) {
    int gid = blockIdx.x * 256 + threadIdx.x;      // over n*128
    if (gid >= n * FEAT) return;
    int i = __builtin_amdgcn_readfirstlane(gid >> 7);  // uniform within wave32
    int f = gid & (FEAT - 1);
    float di = dinv[i];
    AGG[gid] = b[f] + H[gid] * di * di;
}

// ---------------------------------------------------------------------------
// Edge scatter: one wave32 per edge; each lane owns 4 features (float4 gather
// of h[src], scaled by the symmetric norm, 4 global f32 atomics into agg[dst]).
// This kernel is the L2-atomic-bound hot spot: the per-edge metadata
// (src/dst/w/dinv) is wave-uniform, so force it through readfirstlane to get
// one scalar (SMEM) load per wave instead of 32 redundant VMEM loads.
// ---------------------------------------------------------------------------
__global__ void __launch_bounds__(256)
edge_scatter_kernel(const long long* __restrict__ src, const long long* __restrict__ dst,
                    const float* __restrict__ w, const float* __restrict__ dinv,
                    const float* __restrict__ H, float* __restrict__ AGG, int e) {
    int gid  = blockIdx.x * 256 + threadIdx.x;
    int edge = __builtin_amdgcn_readfirstlane(gid >> 5);   // wave-uniform -> SGPR
    if (edge >= e) return;
    int lane = gid & 31;

    long long s = src[edge];
    long long d = dst[edge];
    float nrm = dinv[s] * w[edge] * dinv[d];

    const float4 hv = *(const float4*)(H + (size_t)s * FEAT + lane * 4);
    float* out = AGG + (size_t)d * FEAT + lane * 4;
    atomicAdd(out + 0, hv.x * nrm);
    atomicAdd(out + 1, hv.y * nrm);
    atomicAdd(out + 2, hv.z * nrm);
    atomicAdd(out + 3, hv.w * nrm);
}

// ---------------------------------------------------------------------------
// PReLU + LayerNorm: one wave32 per node, 4 features/lane, __shfl_xor tree
// reductions across the 32 lanes (wave32 is the native width on gfx1250).
// Safe to run in-place (each wave reads then writes only its own row).
// ---------------------------------------------------------------------------
__global__ void __launch_bounds__(256)
prelu_ln_kernel(const float* __restrict__ AGG, const float* __restrict__ prelu_a,
                const float* __restrict__ gamma, const float* __restrict__ beta,
                float* __restrict__ OUT, int n) {
    int wave = threadIdx.x >> 5;
    int lane = threadIdx.x & 31;
    int node = blockIdx.x * 8 + wave;
    if (node >= n) return;

    const float a = prelu_a[0];
    float4 v = *(const float4*)(AGG + (size_t)node * FEAT + lane * 4);
    v.x = (v.x >= 0.0f) ? v.x : a * v.x;
    v.y = (v.y >= 0.0f) ? v.y : a * v.y;
    v.z = (v.z >= 0.0f) ? v.z : a * v.z;
    v.w = (v.w >= 0.0f) ? v.w : a * v.w;

    float s = v.x + v.y + v.z + v.w;
#pragma unroll
    for (int off = 16; off > 0; off >>= 1) s += __shfl_xor(s, off, 32);
    const float mu = s * (1.0f / FEAT);

    float dx = v.x - mu, dy = v.y - mu, dz = v.z - mu, dw = v.w - mu;
    float ss = dx * dx + dy * dy + dz * dz + dw * dw;
#pragma unroll
    for (int off = 16; off > 0; off >>= 1) ss += __shfl_xor(ss, off, 32);
    const float rstd = rsqrtf(ss * (1.0f / FEAT) + EPS);

    const int f = lane * 4;
    float4 o;
    o.x = dx * rstd * gamma[f + 0] + beta[f + 0];
    o.y = dy * rstd * gamma[f + 1] + beta[f + 1];
    o.z = dz * rstd * gamma[f + 2] + beta[f + 2];
    o.w = dw * rstd * gamma[f + 3] + beta[f + 3];
    *(float4*)(OUT + (size_t)node * FEAT + lane * 4) = o;
}

// ---------------------------------------------------------------------------
// Launch: deg once, then two layers. Scratch layout in d_ws:
//   dinv : N floats
//   Hbuf : N*128 floats (GEMM output, gathered by edge scatter)
//   Abuf : N*128 floats (aggregation accumulator / layer-1 activations)
// Layer-2 reuses Hbuf/Abuf after layer-1's consumers complete (stream order).
// ---------------------------------------------------------------------------
extern "C" void kernel_launch(void* const* d_in, const int* in_sizes, int n_in,
                              void* d_out, int out_size, void* d_ws, size_t ws_size,
                              hipStream_t stream) {
    const float*     x     = (const float*)d_in[0];
    const long long* ei    = (const long long*)d_in[1];   // int64 (2,E)
    const float*     ew    = (const float*)d_in[2];
    const float*     W1    = (const float*)d_in[3];
    const float*     b1    = (const float*)d_in[4];
    const float*     W2    = (const float*)d_in[5];
    const float*     b2    = (const float*)d_in[6];
    const float*     pa    = (const float*)d_in[7];
    const float*     gamma = (const float*)d_in[8];
    const float*     beta  = (const float*)d_in[9];

    const int N = in_sizes[0] / FEAT;     // 100000
    const int E = in_sizes[2];            // 1600000
    const long long* src = ei;
    const long long* dst = ei + E;

    char* ws = (char*)d_ws;
    float* dinv = (float*)ws;
    size_t off = ((size_t)N * sizeof(float) + 511) & ~(size_t)511;
    float* Hbuf = (float*)(ws + off);
    off += (size_t)N * FEAT * sizeof(float);
    float* Abuf = (float*)(ws + off);
    float* out  = (float*)d_out;

    const int tN   = (N + 255) / 256;
    const int tE   = (E + 255) / 256;
    const int tNF  = (N * FEAT + 255) / 256;
    const int tE32 = (int)(((long long)E * 32 + 255) / 256);
    const int gGemm = N / 16;             // exact: 6250
    const int gLN   = (N + 7) / 8;

    // Degrees + symmetric normalization (shared by both layers)
    deg_init_kernel<<<tN, 256, 0, stream>>>(dinv, N);
    deg_scatter_kernel<<<tE, 256, 0, stream>>>(dst, ew, dinv, E);
    deg_finalize_kernel<<<tN, 256, 0, stream>>>(dinv, N);

    // ---- Layer 1 ----
    gemm128_wmma_kernel<<<gGemm, 256, 0, stream>>>(x, W1, Hbuf);
    agg_init_kernel<<<tNF, 256, 0, stream>>>(Hbuf, dinv, b1, Abuf, N);
    edge_scatter_kernel<<<tE32, 256, 0, stream>>>(src, dst, ew, dinv, Hbuf, Abuf, E);
    prelu_ln_kernel<<<gLN, 256, 0, stream>>>(Abuf, pa, gamma, beta, Abuf, N); // in-place

    // ---- Layer 2 ----
    gemm128_wmma_kernel<<<gGemm, 256, 0, stream>>>(Abuf, W2, Hbuf);
    agg_init_kernel<<<tNF, 256, 0, stream>>>(Hbuf, dinv, b2, Abuf, N);
    edge_scatter_kernel<<<tE32, 256, 0, stream>>>(src, dst, ew, dinv, Hbuf, Abuf, E);
    prelu_ln_kernel<<<gLN, 256, 0, stream>>>(Abuf, pa, gamma, beta, out, N);
}